// NormalizedHungarianLoss_6459630813758
// MI455X (gfx1250) — compile-verified
//
#include <hip/hip_runtime.h>
#include <stdint.h>

#define N 128
#define LSAP_INF 1e30f

typedef unsigned int uint32x4 __attribute__((ext_vector_type(4)));
typedef int          int32x8  __attribute__((ext_vector_type(8)));
typedef int          int32x4  __attribute__((ext_vector_type(4)));

#if defined(__AMDGCN__) && __has_builtin(__builtin_amdgcn_tensor_load_to_lds) && \
    __has_builtin(__builtin_amdgcn_s_wait_tensorcnt)
#define LSAP_HAS_TDM 1
#else
#define LSAP_HAS_TDM 0
#endif

// One workgroup (128 threads = 4 wave32) solves one 128x128 LSAP instance.
// Cost tile staged into LDS via the Tensor Data Mover; all solver state lives
// in LDS; argmin reductions use wave32 xor-shuffles + 4-way LDS combine.
__global__ __launch_bounds__(N) void lsap_solve_kernel(const float* __restrict__ D,
                                                       float* __restrict__ partial) {
    extern __shared__ float Csh[];                // N*N floats (64 KB) — TDM destination
    __shared__ float u[N], v[N], spc[N];
    __shared__ int   path[N], remaining[N], SR[N], col4row[N], row4col[N];
    __shared__ float s_rval[4], s_red2[4];
    __shared__ int   s_ridx[4];
    __shared__ float s_minVal, s_dmin, s_inv;
    __shared__ int   s_i, s_sink;

    const int tid = threadIdx.x;
    const int b   = blockIdx.x;

    // ---------------- Stage cost tile into LDS ----------------
#if LSAP_HAS_TDM
    if (tid < 32) {
        uint64_t gaddr  = (uint64_t)(uintptr_t)(D + (size_t)b * (N * N));
        uint32_t ldsOff = (uint32_t)(uintptr_t)(&Csh[0]);   // flat addr low 32 bits == LDS offset
        // D# group 0: count=1 (valid), lds_addr, global_addr[56:0], type=2 ("image")
        uint32x4 g0 = { 1u,
                        ldsOff,
                        (uint32_t)gaddr,
                        ((uint32_t)(gaddr >> 32) & 0x01FFFFFFu) | (2u << 30) };
        // D# group 1: data_size=2 (4B); tensor_dim0=tensor_dim1=128;
        // tile_dim0=tile_dim1=128; tensor_dim0_stride=128; no pad/iterate/multicast.
        int32x8 g1 = { (int)0x00020000u,      // workgroup_mask=0, data_size=2
                       (int)(N << 16),        // barrier_addr=0 | tensor_dim0[15:0]
                       (int)(N << 16),        // tensor_dim0[31:16]=0 | tensor_dim1[15:0]
                       (int)(N << 16),        // tensor_dim1[31:16]=0 | tile_dim0
                       N,                     // tile_dim1 | tile_dim2=0
                       N,                     // tensor_dim0_stride[31:0]
                       0, 0 };                // dim0_stride hi / dim1_stride (unused, 2D)
        int32x4 gz4 = { 0, 0, 0, 0 };         // groups 2/3: tile dims 0 => unused
        int32x8 gz8 = { 0, 0, 0, 0, 0, 0, 0, 0 };
        __builtin_amdgcn_tensor_load_to_lds(g0, g1, gz4, gz4, gz8, 0);
    }
    __builtin_amdgcn_s_wait_tensorcnt(0);     // trivial for waves 1..3
#else
    for (int k = tid; k < N * N; k += N)
        Csh[k] = D[(size_t)b * (N * N) + k];
#endif
    __syncthreads();

    // ---------------- Per-sample normalization constants ----------------
    float lmin = LSAP_INF, lmax = -LSAP_INF;
    for (int i = 0; i < N; ++i) {
        float c = Csh[i * N + tid];
        lmin = fminf(lmin, c);
        lmax = fmaxf(lmax, c);
    }
    for (int off = 16; off; off >>= 1) {
        lmin = fminf(lmin, __shfl_xor(lmin, off, 32));
        lmax = fmaxf(lmax, __shfl_xor(lmax, off, 32));
    }
    if ((tid & 31) == 0) { s_rval[tid >> 5] = lmin; s_red2[tid >> 5] = lmax; }
    __syncthreads();
    if (tid == 0) {
        float dmin = fminf(fminf(s_rval[0], s_rval[1]), fminf(s_rval[2], s_rval[3]));
        float dmax = fmaxf(fmaxf(s_red2[0], s_red2[1]), fmaxf(s_red2[2], s_red2[3]));
        s_dmin = dmin;
        s_inv  = 1.0f / (dmax - dmin + 1e-8f);
    }

    // ---------------- Solver state init ----------------
    u[tid] = 0.0f; v[tid] = 0.0f;
    col4row[tid] = -1; row4col[tid] = -1;
    __syncthreads();
    const float dmin = s_dmin, inv = s_inv;

    // ---------------- Jonker-Volgenant: one augmenting path per row ----------------
    for (int cur = 0; cur < N; ++cur) {
        spc[tid] = LSAP_INF; path[tid] = -1; remaining[tid] = 1; SR[tid] = 0;
        if (tid == 0) { s_i = cur; s_minVal = 0.0f; s_sink = -1; }
        __syncthreads();

        while (true) {
            const int   i  = s_i;
            const float mv = s_minVal;
            if (tid == 0) SR[i] = 1;

            float val;
            if (remaining[tid]) {
                float cn = (Csh[i * N + tid] - dmin) * inv;     // normalized cost
                float r  = mv + cn - u[i] - v[tid];
                if (r < spc[tid]) { spc[tid] = r; path[tid] = i; }
                val = spc[tid];
            } else {
                val = LSAP_INF;
            }

            // lexicographic argmin (lowest index on ties, like jnp.argmin)
            float rv = val; int ri = tid;
            for (int off = 16; off; off >>= 1) {
                float ov = __shfl_xor(rv, off, 32);
                int   oi = __shfl_xor(ri, off, 32);
                if (ov < rv || (ov == rv && oi < ri)) { rv = ov; ri = oi; }
            }
            if ((tid & 31) == 0) { s_rval[tid >> 5] = rv; s_ridx[tid >> 5] = ri; }
            __syncthreads();

            if (tid == 0) {
                float bv = s_rval[0]; int bi = s_ridx[0];
#pragma unroll
                for (int w = 1; w < 4; ++w)
                    if (s_rval[w] < bv || (s_rval[w] == bv && s_ridx[w] < bi)) {
                        bv = s_rval[w]; bi = s_ridx[w];
                    }
                s_minVal = bv;
                remaining[bi] = 0;                 // SC[bi] = true
                int owner = row4col[bi];
                if (owner < 0) s_sink = bi; else s_i = owner;
            }
            __syncthreads();
            if (s_sink >= 0) break;
        }

        // ---- dual updates (pre-augment col4row, exactly as scipy _lsap) ----
        const float mv = s_minVal;
        if (tid == cur) {
            u[tid] += mv;
        } else if (SR[tid]) {
            int cj = col4row[tid]; if (cj < 0) cj = 0;
            u[tid] += mv - spc[cj];
        }
        if (!remaining[tid])                        // SC == !remaining
            v[tid] -= mv - spc[tid];
        __syncthreads();

        // ---- augment alternating path back to cur ----
        if (tid == 0) {
            int j = s_sink;
            while (true) {
                int i = path[j];
                row4col[j] = i;
                int jn = col4row[i];
                col4row[i] = j;
                if (i == cur) break;
                j = jn;
            }
        }
        __syncthreads();
    }

    // ---------------- Matched sum over ORIGINAL costs ----------------
    float m = Csh[tid * N + col4row[tid]];
    for (int off = 16; off; off >>= 1) m += __shfl_xor(m, off, 32);
    if ((tid & 31) == 0) s_rval[tid >> 5] = m;
    __syncthreads();
    if (tid == 0)
        partial[b] = s_rval[0] + s_rval[1] + s_rval[2] + s_rval[3];
}

__global__ __launch_bounds__(N) void lsap_finalize_kernel(const float* __restrict__ partial,
                                                          float* __restrict__ out) {
    __shared__ float s_red[4];
    const int tid = threadIdx.x;
    float x = partial[tid];                         // B == 128 partial sums
    for (int off = 16; off; off >>= 1) x += __shfl_xor(x, off, 32);
    if ((tid & 31) == 0) s_red[tid >> 5] = x;
    __syncthreads();
    if (tid == 0)
        out[0] = (s_red[0] + s_red[1] + s_red[2] + s_red[3]) *
                 (1.0f / (128.0f * 128.0f));        // mean over B*n matched entries
}

extern "C" void kernel_launch(void* const* d_in, const int* in_sizes, int n_in,
                              void* d_out, int out_size, void* d_ws, size_t ws_size,
                              hipStream_t stream) {
    (void)in_sizes; (void)n_in; (void)out_size; (void)ws_size;
    const float* D       = (const float*)d_in[0];   // [128, 128, 128] f32
    float*       out     = (float*)d_out;           // scalar f32
    float*       partial = (float*)d_ws;            // 128 floats of scratch

    lsap_solve_kernel<<<128, N, N * N * sizeof(float), stream>>>(D, partial);
    lsap_finalize_kernel<<<1, N, 0, stream>>>(partial, out);
}